// QuantumLayer_23373212024916
// MI455X (gfx1250) — compile-verified
//
#include <hip/hip_runtime.h>

typedef __attribute__((ext_vector_type(2))) float v2f;
typedef __attribute__((ext_vector_type(4))) float v4f;
typedef __attribute__((ext_vector_type(8))) float v8f;

#define DIM 16

// ---------------------------------------------------------------------------
// Kernel 1: build B = (U2*U1)^T  (256 floats into d_ws).
//   U_l = CNOT_RING @ kron(RY(th_l0)..RY(th_l3)),  wire 0 = MSB.
//   CNOT_RING is a basis permutation: U_l[r][c] = K_l[pinv(r)][c].
// Thread t = r*16+c computes M[r][c] = sum_m U2[r][m]*U1[m][c], stores M^T.
// ---------------------------------------------------------------------------
__global__ void build_B_kernel(const float* __restrict__ weights,
                               float* __restrict__ Bmat) {
  const int t = threadIdx.x;   // 0..255
  const int r = t >> 4;
  const int c = t & 15;

  float cs[2][4], sn[2][4];
#pragma unroll
  for (int l = 0; l < 2; ++l)
#pragma unroll
    for (int q = 0; q < 4; ++q) {
      float h = 0.5f * weights[l * 4 + q];
      cs[l][q] = cosf(h);
      sn[l][q] = sinf(h);
    }

  // perm(b): apply CNOT (0,1),(1,2),(2,3),(3,0) on bits (wire0 = MSB); invert.
  int pinv[16];
#pragma unroll
  for (int b = 0; b < 16; ++b) {
    int b0 = (b >> 3) & 1, b1 = (b >> 2) & 1, b2 = (b >> 1) & 1, b3 = b & 1;
    if (b0) b1 ^= 1;
    if (b1) b2 ^= 1;
    if (b2) b3 ^= 1;
    if (b3) b0 ^= 1;
    pinv[(b0 << 3) | (b1 << 2) | (b2 << 1) | b3] = b;
  }

  // kron-of-RY element: K_l[a][b] = prod_w RY(th_lw)[bit_w(a)][bit_w(b)]
  auto kel = [&](int l, int a, int b) -> float {
    float p = 1.0f;
#pragma unroll
    for (int w = 0; w < 4; ++w) {
      int i = (a >> (3 - w)) & 1;
      int j = (b >> (3 - w)) & 1;
      float e = (i == j) ? cs[l][w] : ((i == 0) ? -sn[l][w] : sn[l][w]);
      p *= e;
    }
    return p;
  };

  float m = 0.0f;
  const int pr = pinv[r];
#pragma unroll
  for (int k = 0; k < 16; ++k)
    m += kel(1, pr, k) * kel(0, pinv[k], c);

  Bmat[c * 16 + r] = m;  // store transposed: B[k][n] = M[n][k]
}

// ---------------------------------------------------------------------------
// Kernel 2: persistent wave32 processes a strided sequence of 16-row tiles.
//   Y(16x16) = X_tile(16x16) * M^T   via 4x V_WMMA_F32_16X16X4_F32
//   out[row][i] = (sum_c Y[row][c]^2 * zsign(c,i)) / (sum_c Y[row][c]^2)
//   (M orthogonal -> denominator == ||x_row||^2; normalization folded in)
// LDS exchange is strictly intra-wave: LDS ops of one wave are processed
// in order by the LDS unit (DScnt in-order), so no fence/barrier instruction
// is needed — only a compile-time scheduling barrier.
// ---------------------------------------------------------------------------
#define ROWSTRIDE 20  // floats; 80B rows -> 16B aligned, conflict-free banks
#define WAVES_PER_BLOCK 8

__global__ __launch_bounds__(256) void qexp_kernel(const float* __restrict__ x,
                                                   const float* __restrict__ Bmat,
                                                   float* __restrict__ out,
                                                   long long ntiles) {
  __shared__ float lds[WAVES_PER_BLOCK][16 * ROWSTRIDE];

  const int lane = threadIdx.x & 31;
  const int wave = threadIdx.x >> 5;
  const int half = lane >> 4;   // 0: lanes 0-15, 1: lanes 16-31
  const int lr   = lane & 15;
  float* myLds = &lds[wave][0];

  const long long stride = (long long)gridDim.x * WAVES_PER_BLOCK;  // waves total
  long long t = (long long)blockIdx.x * WAVES_PER_BLOCK + wave;
  if (t >= ntiles) return;      // wave-uniform

  // B operand (M^T), hoisted: per k-step k, vgpr0 = K rows {4k, 4k+2},
  // vgpr1 = {4k+1, 4k+3}. Stays cached in L2 (x streams past it as NT).
  v2f b0, b1, b2, b3;
  {
    const float* bp = Bmat + lr;
    const int ro = 2 * half;
    b0.x = bp[(0  + ro) * 16]; b0.y = bp[(1  + ro) * 16];
    b1.x = bp[(4  + ro) * 16]; b1.y = bp[(5  + ro) * 16];
    b2.x = bp[(8  + ro) * 16]; b2.y = bp[(9  + ro) * 16];
    b3.x = bp[(12 + ro) * 16]; b3.y = bp[(13 + ro) * 16];
  }

  // Per-lane pointers for tile t. A operand layout for 16x16x4 f32:
  // lane (half,lr) holds X[tileRow lr][4k + 2*half + {0,1}] at k-step k.
  const float* xp = x + t * 256 + lr * DIM + 2 * half;
  float*       op = out + t * 64 + lr * 4;

  // Preload first tile (non-temporal: x is stream-once, larger than L2).
  v2f a0 = __builtin_nontemporal_load((const v2f*)(xp + 0));
  v2f a1 = __builtin_nontemporal_load((const v2f*)(xp + 4));
  v2f a2 = __builtin_nontemporal_load((const v2f*)(xp + 8));
  v2f a3 = __builtin_nontemporal_load((const v2f*)(xp + 12));

  while (true) {
    const long long tn = t + stride;
    const bool more = (tn < ntiles);

    // --- matmul: Y = X * M^T ---
    v8f acc = {};
    acc = __builtin_amdgcn_wmma_f32_16x16x4_f32(false, a0, false, b0, (short)0, acc, false, false);
    acc = __builtin_amdgcn_wmma_f32_16x16x4_f32(false, a1, false, b1, (short)0, acc, false, false);
    acc = __builtin_amdgcn_wmma_f32_16x16x4_f32(false, a2, false, b2, (short)0, acc, false, false);
    acc = __builtin_amdgcn_wmma_f32_16x16x4_f32(false, a3, false, b3, (short)0, acc, false, false);

    // --- prefetch next tile unconditionally (clamped pointer on the last
    //     iteration: re-reads the current tile, result unused) ---
    const float* xpn = more ? (xp + stride * 256) : xp;
    a0 = __builtin_nontemporal_load((const v2f*)(xpn + 0));
    a1 = __builtin_nontemporal_load((const v2f*)(xpn + 4));
    a2 = __builtin_nontemporal_load((const v2f*)(xpn + 8));
    a3 = __builtin_nontemporal_load((const v2f*)(xpn + 12));

    // --- D layout: vgpr j, lane -> row (j + 8*half), component lr ---
#pragma unroll
    for (int j = 0; j < 8; ++j) {
      float y = acc[j];
      myLds[(j + 8 * half) * ROWSTRIDE + lr] = y * y;
    }

    // intra-wave LDS RAW: hardware processes same-wave LDS ops in order;
    // only forbid compile-time reordering (emits no instruction).
    __builtin_amdgcn_wave_barrier();

    // --- lane lr reduces row lr: denom + 4 signed sums (ZSIGNS pattern) ---
    float den = 0.f, e0 = 0.f, e1 = 0.f, e2 = 0.f, e3 = 0.f;
#pragma unroll
    for (int c = 0; c < 16; ++c) {
      float p = myLds[lr * ROWSTRIDE + c];
      den += p;
      e0 += (c & 8) ? -p : p;   // qubit 0 = MSB
      e1 += (c & 4) ? -p : p;
      e2 += (c & 2) ? -p : p;
      e3 += (c & 1) ? -p : p;
    }

    if (half == 0) {
      float inv = 1.0f / den;
      v4f o = {e0 * inv, e1 * inv, e2 * inv, e3 * inv};
      __builtin_nontemporal_store(o, (v4f*)op);  // 256B coalesced per tile
    }

    // WAR guard: next iteration's LDS stores must not be scheduled above
    // this iteration's LDS reads (hardware is in-order; compiler fence only).
    __builtin_amdgcn_wave_barrier();

    if (!more) break;
    t = tn;
    xp = xpn;
    op += stride * 64;
  }
}

// ---------------------------------------------------------------------------
extern "C" void kernel_launch(void* const* d_in, const int* in_sizes, int n_in,
                              void* d_out, int out_size, void* d_ws, size_t ws_size,
                              hipStream_t stream) {
  const float* x = (const float*)d_in[0];     // [B,16] f32
  const float* w = (const float*)d_in[1];     // [2,4]  f32
  float* out  = (float*)d_out;                // [B,4]  f32
  float* Bmat = (float*)d_ws;                 // 256 floats scratch

  build_B_kernel<<<1, 256, 0, stream>>>(w, Bmat);

  long long nrows  = (long long)in_sizes[0] / DIM;
  long long ntiles = nrows / 16;              // BATCH is a multiple of 16

  long long maxBlocks = (ntiles + WAVES_PER_BLOCK - 1) / WAVES_PER_BLOCK;
  int blocks = (int)(maxBlocks < 4096 ? maxBlocks : 4096);  // persistent waves
  qexp_kernel<<<blocks, 256, 0, stream>>>(x, Bmat, out, ntiles);
}